// ScatterND_1288490189236
// MI455X (gfx1250) — compile-verified
//
#include <hip/hip_runtime.h>

// ScatterND (k=1) on MI455X: pure bandwidth problem (0 FLOPs, ~520 MB traffic
// -> ~22 us at 23.3 TB/s HBM). Single-pass plan: build a row->update table
// (4 MB, stays hot in the 192 MB L2), then one streaming pass that writes every
// output row exactly once, sourcing it from `data` or `updates`. Bulk traffic
// is global_load_b128 / global_store_b128 with TH=NT so the 512 MB stream does
// not evict the table. Hot kernels use shift/mask addressing (no int divides).

typedef float v4f __attribute__((ext_vector_type(4)));

// ---- table path ------------------------------------------------------------

__global__ void init_table_kernel(int* __restrict__ table, int rows) {
    int i = blockIdx.x * blockDim.x + threadIdx.x;
    if (i < rows) table[i] = -1;          // RT store: keep table in L2
}

__global__ void mark_rows_kernel(const int* __restrict__ indices,
                                 int* __restrict__ table, int nupd) {
    int i = blockIdx.x * blockDim.x + threadIdx.x;
    if (i < nupd) table[indices[i]] = i;  // indices unique per ONNX ScatterND
}

// One thread per float4; vec_per_row is a power of two -> row/lane via
// shift/mask (2 VALU ops). 16 consecutive lanes share one row, so the table
// load coalesces to an L2-hit broadcast.
__global__ void emit_rows_v4_pow2_kernel(const v4f* __restrict__ data,
                                         const v4f* __restrict__ updates,
                                         const int* __restrict__ table,
                                         v4f* __restrict__ out,
                                         unsigned total_v4, unsigned shift,
                                         unsigned mask) {
    unsigned t = blockIdx.x * blockDim.x + threadIdx.x;
    if (t >= total_v4) return;
    unsigned row  = t >> shift;
    unsigned lane = t & mask;
    int s = table[row];
    const v4f* src = (s >= 0) ? updates + ((unsigned)s << shift) + lane
                              : data + t;
    v4f v = __builtin_nontemporal_load(src);   // global_load_b128 th:TH_LOAD_NT
    __builtin_nontemporal_store(v, out + t);   // global_store_b128 th:TH_STORE_NT
}

// Generic (non-pow2 cols) variant: single 32-bit divide.
__global__ void emit_rows_v4_gen_kernel(const v4f* __restrict__ data,
                                        const v4f* __restrict__ updates,
                                        const int* __restrict__ table,
                                        v4f* __restrict__ out,
                                        unsigned total_v4, unsigned vpr) {
    unsigned t = blockIdx.x * blockDim.x + threadIdx.x;
    if (t >= total_v4) return;
    unsigned row  = t / vpr;
    unsigned lane = t - row * vpr;
    int s = table[row];
    const v4f* src = (s >= 0) ? updates + (unsigned)s * vpr + lane
                              : data + t;
    v4f v = __builtin_nontemporal_load(src);
    __builtin_nontemporal_store(v, out + t);
}

// ---- fallback path (table does not fit): copy then scatter -----------------

__global__ void copy_v4_kernel(const v4f* __restrict__ in, v4f* __restrict__ out,
                               long long n) {
    long long t = (long long)blockIdx.x * blockDim.x + threadIdx.x;
    if (t >= n) return;
    __builtin_nontemporal_store(__builtin_nontemporal_load(in + t), out + t);
}

__global__ void scatter_v4_kernel(const v4f* __restrict__ updates,
                                  const int* __restrict__ indices,
                                  v4f* __restrict__ out,
                                  unsigned total_v4, unsigned vpr) {
    unsigned t = blockIdx.x * blockDim.x + threadIdx.x;
    if (t >= total_v4) return;
    unsigned u    = t / vpr;
    unsigned lane = t - u * vpr;
    long long r = indices[u];
    v4f v = __builtin_nontemporal_load(updates + t);
    __builtin_nontemporal_store(v, out + r * vpr + lane);
}

// Scalar fallback for cols % 4 != 0 (never hit in this workload).
__global__ void copy_scalar_kernel(const float* __restrict__ in,
                                   float* __restrict__ out, long long n) {
    long long t = (long long)blockIdx.x * blockDim.x + threadIdx.x;
    if (t < n)
        __builtin_nontemporal_store(__builtin_nontemporal_load(in + t), out + t);
}

__global__ void scatter_scalar_kernel(const float* __restrict__ updates,
                                      const int* __restrict__ indices,
                                      float* __restrict__ out,
                                      unsigned total, unsigned cols) {
    unsigned t = blockIdx.x * blockDim.x + threadIdx.x;
    if (t >= total) return;
    unsigned u = t / cols;
    unsigned c = t - u * cols;
    long long r = indices[u];
    out[r * cols + c] = updates[t];
}

// ---- launcher --------------------------------------------------------------

extern "C" void kernel_launch(void* const* d_in, const int* in_sizes, int n_in,
                              void* d_out, int out_size, void* d_ws, size_t ws_size,
                              hipStream_t stream) {
    const float* data    = (const float*)d_in[0];
    const int*   indices = (const int*)d_in[1];
    const float* updates = (const float*)d_in[2];
    float*       out     = (float*)d_out;

    const long long data_n = (long long)in_sizes[0];   // 64,000,000
    const int       nupd   = in_sizes[1];              // 100,000
    const long long upd_n  = (long long)in_sizes[2];   // 6,400,000
    const int       cols   = (int)(upd_n / nupd);      // 64
    const long long rows   = data_n / cols;            // 1,000,000

    const int BLK = 256;  // 8 wave32 waves per block

    const bool      vec_ok   = (cols % 4) == 0;
    const int       vpr      = vec_ok ? cols / 4 : 0;  // 16
    const long long total_v4 = vec_ok ? rows * vpr : 0;
    const bool table_ok = vec_ok &&
                          ws_size >= (size_t)rows * sizeof(int) &&
                          total_v4 <= 0x7fffffffLL;

    if (table_ok) {
        int* table = (int*)d_ws;
        unsigned gb0 = (unsigned)((rows + BLK - 1) / BLK);
        init_table_kernel<<<gb0, BLK, 0, stream>>>(table, (int)rows);
        unsigned gb1 = (unsigned)((nupd + BLK - 1) / BLK);
        mark_rows_kernel<<<gb1, BLK, 0, stream>>>(indices, table, nupd);

        unsigned tv4 = (unsigned)total_v4;
        unsigned gb2 = (tv4 + BLK - 1) / BLK;
        if ((vpr & (vpr - 1)) == 0) {              // pow2 vec_per_row (cols=64)
            unsigned shift = (unsigned)__builtin_ctz((unsigned)vpr);
            emit_rows_v4_pow2_kernel<<<gb2, BLK, 0, stream>>>(
                (const v4f*)data, (const v4f*)updates, table, (v4f*)out,
                tv4, shift, (unsigned)vpr - 1u);
        } else {
            emit_rows_v4_gen_kernel<<<gb2, BLK, 0, stream>>>(
                (const v4f*)data, (const v4f*)updates, table, (v4f*)out,
                tv4, (unsigned)vpr);
        }
    } else if (vec_ok && total_v4 <= 0x7fffffffLL) {
        long long n_v4 = data_n / 4;
        unsigned gbc = (unsigned)((n_v4 + BLK - 1) / BLK);
        copy_v4_kernel<<<gbc, BLK, 0, stream>>>((const v4f*)data, (v4f*)out, n_v4);
        unsigned sv4 = (unsigned)((long long)nupd * vpr);
        unsigned gbs = (sv4 + BLK - 1) / BLK;
        scatter_v4_kernel<<<gbs, BLK, 0, stream>>>(
            (const v4f*)updates, indices, (v4f*)out, sv4, (unsigned)vpr);
    } else {
        unsigned gbc = (unsigned)((data_n + BLK - 1) / BLK);
        copy_scalar_kernel<<<gbc, BLK, 0, stream>>>(data, out, data_n);
        unsigned gbs = (unsigned)((upd_n + BLK - 1) / BLK);
        scatter_scalar_kernel<<<gbs, BLK, 0, stream>>>(
            updates, indices, out, (unsigned)upd_n, (unsigned)cols);
    }
}